// NeuralControlCritic_19877108645945
// MI455X (gfx1250) — compile-verified
//
#include <hip/hip_runtime.h>
#include <hip/hip_bf16.h>

// ---------------------------------------------------------------------------
// NeuralControlCritic forward on gfx1250 (CDNA5, wave32, WMMA f32 16x16x4)
//
// Weight-streaming bound (~580 MB f32 weights @ 23.3 TB/s ~= 25us). All GEMMs
// run through v_wmma_f32_16x16x4_f32. A panels are double-buffered in LDS via
// async global->LDS DMA (ASYNCcnt, in-order completion -> s_wait_asynccnt 4
// overlaps the next prefetch with compute). W is streamed with non-temporal
// loads (used once; keeps L2 for reused activations). Skinny GEMMs use
// deterministic split-K so every weight-streaming launch spreads over enough
// workgroups to saturate HBM.
// ---------------------------------------------------------------------------

typedef __attribute__((ext_vector_type(2))) float v2f;
typedef __attribute__((ext_vector_type(8))) float v8f;
typedef __attribute__((ext_vector_type(4))) int   v4i;

#define B_   32
#define SEQ  32
#define DM   128
#define DI   256
#define DS_  64
#define NH   4
#define HD   64
#define DPJ  644   // 2*DI + 2*DS + NH
#define XBCW 384   // DI + 2*DS

#define CK       128          // K-chunk staged in LDS
#define LDST     132          // padded LDS row stride (floats), 16B multiple
#define AS1      __attribute__((address_space(1)))
#define AS3      __attribute__((address_space(3)))

#if defined(__gfx1250__) && __has_builtin(__builtin_amdgcn_global_load_async_to_lds_b128)
#define USE_ASYNC_LDS 1
#else
#define USE_ASYNC_LDS 0
#endif

// Stage one 32xCK A chunk into an LDS buffer (4 x b128 per thread).
__device__ __forceinline__ void stage_chunk(const float* __restrict__ A, int K,
                                            int m0, int k0, int tid, float* buf)
{
#pragma unroll
    for (int t = 0; t < 4; ++t) {
        const int idx = tid + t * 256;            // 1024 float4 = 32x(CK/4)
        const int row = idx >> 5;
        const int seg = idx & 31;
        const float* gp = A + (size_t)(m0 + row) * K + k0 + seg * 4;
        float* lp = buf + row * LDST + seg * 4;
#if USE_ASYNC_LDS
        __builtin_amdgcn_global_load_async_to_lds_b128(
            (AS1 v4i*)gp, (AS3 v4i*)lp, 0, 0);
#else
        *(float4*)lp = *(const float4*)gp;
#endif
    }
}

// ---------------------------------------------------------------------------
// Generic GEMM slice: Cout = A[.., kb:kb+kslice] @ W[kb:kb+kslice, :]
// Block: 256 threads (8 waves). Wave w -> 32x32 tile. Columns clamped (not
// predicated): no EXEC divergence in the K loop. partial=1: raw slice partial
// to C + blockIdx.z*M*N. partial=0: fused bias + optional ReLU.
// W addressing: uniform SGPR base (k-row) + loop-invariant lane offset
// -> saddr + scaled-voffset loads, no per-iteration vector address math.
// ---------------------------------------------------------------------------
__global__ __launch_bounds__(256)
void gemm_wmma_f32(const float* __restrict__ A, const float* __restrict__ W,
                   const float* __restrict__ bias, float* __restrict__ C,
                   int M, int N, int K, int kslice, int partial, int relu)
{
    __shared__ float sA[2][32 * LDST];

    const int tid   = threadIdx.x;
    const int lane  = tid & 31;
    const int wave  = tid >> 5;
    const int m0    = blockIdx.y * 32;
    const int n0    = (blockIdx.x * 8 + wave) * 32;   // 32 columns per wave
    const int col   = lane & 15;
    const int khalf = (lane >> 4) << 1;               // 0 | 2
    const int rA    = lane & 15;

    const int nc0 = n0 + col;
    const int nc1 = n0 + 16 + col;
    const int cc0 = nc0 < N ? nc0 : N - 1;            // clamped (no EXEC div.)
    const int cc1 = nc1 < N ? nc1 : N - 1;

    // loop-invariant per-lane W offsets (elements); uniform part is in Wk
    const int l0a = cc0 + khalf * N;
    const int l0b = l0a + N;
    const int l1a = cc1 + khalf * N;
    const int l1b = l1a + N;

    const int kb = blockIdx.z * kslice;
    const int ke = kb + kslice;

    v8f c00 = {}, c10 = {}, c01 = {}, c11 = {};

    stage_chunk(A, K, m0, kb, tid, sA[0]);            // prologue prefetch

    int buf = 0;
    for (int k0 = kb; k0 < ke; k0 += CK, buf ^= 1) {
        const bool more = (k0 + CK) < ke;             // wave-uniform
        if (more) stage_chunk(A, K, m0, k0 + CK, tid, sA[buf ^ 1]);
#if USE_ASYNC_LDS
        if (more) asm volatile("s_wait_asynccnt 0x4" ::: "memory");
        else      asm volatile("s_wait_asynccnt 0x0" ::: "memory");
#endif
        __syncthreads();                              // chunk k0 visible

        const float* sAb = sA[buf];
#pragma unroll 4
        for (int kk = 0; kk < CK; kk += 4) {
            const int ka = kk + khalf;
            const float* Wk = W + (size_t)(k0 + kk) * N;   // uniform (SGPR)
            v2f a0, a1, b0, b1;
            a0.x = sAb[rA * LDST + ka];
            a0.y = sAb[rA * LDST + ka + 1];
            a1.x = sAb[(rA + 16) * LDST + ka];
            a1.y = sAb[(rA + 16) * LDST + ka + 1];
            b0.x = __builtin_nontemporal_load(Wk + l0a);
            b0.y = __builtin_nontemporal_load(Wk + l0b);
            b1.x = __builtin_nontemporal_load(Wk + l1a);
            b1.y = __builtin_nontemporal_load(Wk + l1b);
            c00 = __builtin_amdgcn_wmma_f32_16x16x4_f32(false, a0, false, b0,
                                                        (short)0, c00, false, false);
            c10 = __builtin_amdgcn_wmma_f32_16x16x4_f32(false, a1, false, b0,
                                                        (short)0, c10, false, false);
            c01 = __builtin_amdgcn_wmma_f32_16x16x4_f32(false, a0, false, b1,
                                                        (short)0, c01, false, false);
            c11 = __builtin_amdgcn_wmma_f32_16x16x4_f32(false, a1, false, b1,
                                                        (short)0, c11, false, false);
        }
        __syncthreads();                              // protect buf^1 rewrite
    }

    float* Cout = partial ? (C + (size_t)blockIdx.z * M * N) : C;
    const bool ok0 = nc0 < N;
    const bool ok1 = nc1 < N;
    const float bv0 = (!partial && bias && ok0) ? bias[nc0] : 0.f;
    const float bv1 = (!partial && bias && ok1) ? bias[nc1] : 0.f;
    const int rbase = (lane >> 4) << 3;               // 0 | 8
#pragma unroll
    for (int v = 0; v < 8; ++v) {
        const int r0 = m0 + rbase + v;
        const int r1 = r0 + 16;
        float f00 = c00[v] + bv0, f10 = c10[v] + bv0;
        float f01 = c01[v] + bv1, f11 = c11[v] + bv1;
        if (relu && !partial) {
            f00 = fmaxf(f00, 0.f); f10 = fmaxf(f10, 0.f);
            f01 = fmaxf(f01, 0.f); f11 = fmaxf(f11, 0.f);
        }
        if (ok0) {
            Cout[(size_t)r0 * N + nc0] = f00;
            Cout[(size_t)r1 * N + nc0] = f10;
        }
        if (ok1) {
            Cout[(size_t)r0 * N + nc1] = f01;
            Cout[(size_t)r1 * N + nc1] = f11;
        }
    }
}

// ---------------------------------------------------------------------------
// Deterministic split-K reduction: C[i] = act(sum_z CP[z][i] + bias[i%N]).
// Fixed summation order -> bitwise deterministic.
// ---------------------------------------------------------------------------
__global__ __launch_bounds__(256)
void splitk_reduce(const float* __restrict__ CP, const float* __restrict__ bias,
                   float* __restrict__ C, int total, int N, int SK, int relu)
{
    const int i = blockIdx.x * 256 + threadIdx.x;
    if (i >= total) return;
    float acc = bias ? bias[i % N] : 0.f;
    for (int z = 0; z < SK; ++z) acc += CP[(size_t)z * total + i];
    if (relu) acc = fmaxf(acc, 0.f);
    C[i] = acc;
}

// ---------------------------------------------------------------------------
// Depthwise causal conv (DC=4) over xBC channels + SiLU.
// ZX rows are [z(256) | xBC(384) | dt(4)], stride DPJ.
// ---------------------------------------------------------------------------
__global__ __launch_bounds__(256)
void conv_silu(const float* __restrict__ ZX, const float* __restrict__ cw,
               const float* __restrict__ cb, float* __restrict__ XB)
{
    const int idx = blockIdx.x * blockDim.x + threadIdx.x;
    if (idx >= B_ * SEQ * XBCW) return;
    const int c   = idx % XBCW;
    const int row = idx / XBCW;          // b*SEQ + l
    const int l   = row & (SEQ - 1);
    const int b   = row >> 5;
    float acc = cb[c];
#pragma unroll
    for (int k = 0; k < 4; ++k) {
        const int lp = l + k - 3;
        if (lp >= 0)
            acc += cw[k * XBCW + c] * ZX[(size_t)(b * SEQ + lp) * DPJ + DI + c];
    }
    const float s = 1.f / (1.f + __expf(-acc));
    XB[idx] = acc * s;
}

// ---------------------------------------------------------------------------
// Mamba2 selective scan. One block per (b, head). 256 threads: t = p*4 + q,
// p in [0,64) owns state row h[p][:], q in [0,4) owns 16 of its 64 columns.
// C-contraction reduced over the quad with __shfl_xor (wave32).
// ---------------------------------------------------------------------------
__global__ __launch_bounds__(256)
void mamba_scan(const float* __restrict__ ZX, const float* __restrict__ XB,
                const float* __restrict__ A_log, const float* __restrict__ Dp,
                const float* __restrict__ dt_bias, float* __restrict__ YR)
{
    __shared__ float sBC[128];           // [0..63]=B_t, [64..127]=C_t
    const int b = blockIdx.x >> 2;
    const int h = blockIdx.x & 3;
    const int t = threadIdx.x;
    const int p = t >> 2;
    const int q = t & 3;

    const float Ah  = -__expf(A_log[h]);
    const float Dh  = Dp[h];
    const float dtb = dt_bias[h];

    float s[16];
#pragma unroll
    for (int j = 0; j < 16; ++j) s[j] = 0.f;

    for (int l = 0; l < SEQ; ++l) {
        const size_t row = (size_t)(b * SEQ + l);
        if (t < 128) sBC[t] = XB[row * XBCW + DI + t];
        __syncthreads();

        const float dtraw = ZX[row * DPJ + 2 * DI + 2 * DS_ + h] + dtb;
        const float dt = (dtraw > 20.f) ? dtraw : __logf(1.f + __expf(dtraw));
        const float dA = __expf(dt * Ah);
        const float xv = XB[row * XBCW + h * HD + p];
        const float dtx = dt * xv;

        float acc = 0.f;
#pragma unroll
        for (int j = 0; j < 16; ++j) {
            const int n = q * 16 + j;
            s[j] = dA * s[j] + dtx * sBC[n];
            acc += s[j] * sBC[64 + n];
        }
        acc += __shfl_xor(acc, 1, 32);
        acc += __shfl_xor(acc, 2, 32);
        if (q == 0) YR[row * DI + h * HD + p] = acc + Dh * xv;
        __syncthreads();
    }
}

// ---------------------------------------------------------------------------
// y = y * silu(z); y *= rsqrt(mean(y^2) + 1e-5) * norm_w. One block per row.
// ---------------------------------------------------------------------------
__global__ __launch_bounds__(256)
void gated_rmsnorm(const float* __restrict__ ZX, const float* __restrict__ YR,
                   const float* __restrict__ nw, float* __restrict__ YN)
{
    __shared__ float red[8];
    const size_t row = blockIdx.x;
    const int c = threadIdx.x;
    const float z = ZX[row * DPJ + c];
    const float g = z / (1.f + __expf(-z));
    const float y = YR[row * DI + c] * g;

    float ss = y * y;
#pragma unroll
    for (int o = 1; o < 32; o <<= 1) ss += __shfl_xor(ss, o, 32);
    if ((c & 31) == 0) red[c >> 5] = ss;
    __syncthreads();
    float tot = 0.f;
#pragma unroll
    for (int w = 0; w < 8; ++w) tot += red[w];
    const float scale = rsqrtf(tot / (float)DI + 1e-5f);
    YN[row * DI + c] = y * scale * nw[c];
}

// ---------------------------------------------------------------------------
// SK>1: per-slice partials into CP, then fixed-order reduce (+bias, +relu).
// SK==1: fused epilogue directly in the GEMM. K/SK must be a multiple of CK.
// ---------------------------------------------------------------------------
static inline void run_gemm(const float* A, const float* W, const float* bias,
                            float* C, float* CP, int M, int N, int K,
                            int relu, int SK, hipStream_t s)
{
    dim3 grid((N + 255) / 256, M / 32, SK);
    if (SK == 1) {
        gemm_wmma_f32<<<grid, 256, 0, s>>>(A, W, bias, C, M, N, K, K, 0, relu);
    } else {
        gemm_wmma_f32<<<grid, 256, 0, s>>>(A, W, nullptr, CP, M, N, K,
                                           K / SK, 1, 0);
        const int total = M * N;
        splitk_reduce<<<(total + 255) / 256, 256, 0, s>>>(CP, bias, C, total,
                                                          N, SK, relu);
    }
}

extern "C" void kernel_launch(void* const* d_in, const int* in_sizes, int n_in,
                              void* d_out, int out_size, void* d_ws, size_t ws_size,
                              hipStream_t stream)
{
    // Pytree-flattened input order (dict keys sorted, lists in order):
    //   [0] x
    //   params  (base 1):  mamba0{A_log,D,conv_b,conv_w,dt_bias,norm_w,w_in,w_out}
    //                      mamba1{...}  (+8)
    //                      mlp_in {b1,b2,w1,w2} (+16..19)
    //                      mlp_out{b1,b2,w1,w2} (+20..23)
    //   ema_params (base 25): same layout.
    (void)in_sizes; (void)n_in; (void)out_size; (void)ws_size;

    const float* x = (const float*)d_in[0];
    float* ws = (float*)d_ws;

    float* H1 = ws;                 // 32 x 16384
    float* H  = H1 + 32 * 16384;    // 32 x 4096  == (1024 x 128)
    float* ZX = H  + 32 * 4096;     // 1024 x 644
    float* XB = ZX + 1024 * DPJ;    // 1024 x 384
    float* YR = XB + 1024 * XBCW;   // 1024 x 256
    float* YN = YR + 1024 * DI;     // 1024 x 256
    float* T  = YN + 1024 * DI;     // 32 x 128
    float* CP = T  + 32 * 128;      // split-K partials, up to 2M floats

    for (int crit = 0; crit < 2; ++crit) {
        const int base = 1 + crit * 24;
        auto P = [&](int i) { return (const float*)d_in[base + i]; };
        float* outp = (float*)d_out + crit * B_ * 8;

        // mlp_in: 32x512 @ 512x16384 (SK=4 -> 256 blocks),
        //         32x16384 @ 16384x4096 (SK=16 -> 256 blocks)
        run_gemm(x,  P(18), P(16), H1, CP, 32, 16384, 512,   1, 4,  stream);
        run_gemm(H1, P(19), P(17), H,  CP, 32, 4096,  16384, 0, 16, stream);

        // two mamba2 layers
        for (int m = 0; m < 2; ++m) {
            const int o = m * 8;
            run_gemm(H, P(6 + o), nullptr, ZX, CP, 1024, DPJ, DM, 0, 1, stream);
            conv_silu<<<(B_ * SEQ * XBCW + 255) / 256, 256, 0, stream>>>(
                ZX, P(3 + o), P(2 + o), XB);
            mamba_scan<<<B_ * NH, 256, 0, stream>>>(
                ZX, XB, P(0 + o), P(1 + o), P(4 + o), YR);
            gated_rmsnorm<<<B_ * SEQ, 256, 0, stream>>>(ZX, YR, P(5 + o), YN);
            run_gemm(YN, P(7 + o), nullptr, H, CP, 1024, DM, DI, 0, 2, stream);
        }

        // mlp_out: 32x4096 @ 4096x128 (SK=32 -> 32 blocks), then 128->8
        run_gemm(H, P(22), P(20), T,    CP, 32, 128, 4096, 1, 32, stream);
        run_gemm(T, P(23), P(21), outp, CP, 32, 8,   128,  0, 1,  stream);
    }
}